// EpisodicMemoryStore_47004122088036
// MI455X (gfx1250) — compile-verified
//
#include <hip/hip_runtime.h>
#include <hip/hip_bf16.h>
#include <math.h>

#define DD   512
#define HH   8
#define DHD  64
#define CAPN 200000
#define NSTRIPE 125      // 125 * 1600 = 200000
#define ROWS_PER_STRIPE 1600

typedef __attribute__((ext_vector_type(2))) float v2f;
typedef __attribute__((ext_vector_type(8))) float v8f;

// ---------------------------------------------------------------- prep:
// q = Wq @ query + bq ; qk[h][j] = 0.125 * sum_d q[h*64+d] * Wk[h*64+d][j]
// qconst[h] = 0.125 * q_h . bk_h
__global__ void k_prep(const float* __restrict__ query,
                       const float* __restrict__ inW,
                       const float* __restrict__ inB,
                       float* __restrict__ qk, float* __restrict__ qconst) {
  __shared__ float qs[DD];
  int tid = threadIdx.x;                     // 512 threads
  const float* wr = inW + (size_t)tid * DD;  // Wq row tid
  float acc = inB[tid];
  for (int j = 0; j < DD; ++j) acc += wr[j] * query[j];
  qs[tid] = acc;
  __syncthreads();
  for (int e = tid; e < HH * DD; e += DD) {
    int h = e >> 9, j = e & (DD - 1);
    const float* wk = inW + (size_t)(DD + h * DHD) * DD + j;  // Wk col j
    float s = 0.f;
    for (int d = 0; d < DHD; ++d) s += qs[h * DHD + d] * wk[(size_t)d * DD];
    qk[e] = 0.125f * s;                      // fold 1/sqrt(64)
  }
  if (tid < HH) {
    float s = 0.f;
    for (int d = 0; d < DHD; ++d) s += qs[tid * DHD + d] * inB[DD + tid * DHD + d];
    qconst[tid] = 0.125f * s;
  }
}

// ---------------------------------------------------------------- pass 1:
// scores[n][h] = M[n] . qk[h] + qconst[h]  via V_WMMA_F32_16X16X4_F32,
// plus per-row squared norms. One 16-row tile per wave. B operand is staged
// in LDS zero-padded to 16 "head" columns so the inner loop is branch-free.
__global__ void k_scores(const float* __restrict__ M,
                         const float* __restrict__ qk,
                         const float* __restrict__ qconst,
                         float* __restrict__ scores,
                         float* __restrict__ sqnorm) {
  __shared__ float qks[16 * DD];             // rows 8..15 zero
  __shared__ float qcs[HH];
  int tid = threadIdx.x;                     // 256 threads = 8 waves
  for (int e = tid; e < HH * DD; e += 256) qks[e] = qk[e];
  for (int e = HH * DD + tid; e < 16 * DD; e += 256) qks[e] = 0.f;
  if (tid < HH) qcs[tid] = qconst[tid];
  __syncthreads();
  int wave = tid >> 5, lane = tid & 31;
  int tile = blockIdx.x * 8 + wave;          // wave-uniform
  if (tile * 16 >= CAPN) return;             // uniform exit: EXEC stays full
  int n0 = tile * 16;
  int m  = lane & 15;
  int ks = (lane < 16) ? 0 : 2;
  const float* row = M + (size_t)(n0 + m) * DD + ks;
  const float* bp  = qks + m * DD + ks;      // B column = head (8..15 -> zeros)
  v8f c = {};
  float sq = 0.f;
#pragma unroll 4
  for (int k0 = 0; k0 < DD; k0 += 4) {
    v2f a = *(const v2f*)(row + k0);         // global_load_b64
    v2f b = *(const v2f*)(bp + k0);          // ds_load_b64, no predication
    sq += a.x * a.x + a.y * a.y;
    c = __builtin_amdgcn_wmma_f32_16x16x4_f32(false, a, false, b,
                                              (short)0, c, false, false);
  }
  // each row's sumsq is split across lane and lane^16
  sq += __shfl_xor(sq, 16, 32);
  if (lane < 16) sqnorm[n0 + m] = sq;
  if (m < HH) {
#pragma unroll
    for (int r = 0; r < 8; ++r) {
      int mrow = (lane < 16) ? r : r + 8;    // D layout: VGPR r -> row r / r+8
      scores[(size_t)(n0 + mrow) * HH + m] = c[r] + qcs[m];
    }
  }
}

// ---------------------------------------------------------------- softmax stats
__global__ void k_stats(const float* __restrict__ scores,
                        float* __restrict__ hmax, float* __restrict__ hsuminv) {
  int h = blockIdx.x, tid = threadIdx.x;     // 8 blocks x 256 threads
  __shared__ float red[256];
  float mx = -3.4e38f;
  for (int n = tid; n < CAPN; n += 256) mx = fmaxf(mx, scores[(size_t)n * HH + h]);
  red[tid] = mx; __syncthreads();
  for (int o = 128; o > 0; o >>= 1) { if (tid < o) red[tid] = fmaxf(red[tid], red[tid + o]); __syncthreads(); }
  float mh = red[0]; __syncthreads();
  float s = 0.f;
  for (int n = tid; n < CAPN; n += 256) s += __expf(scores[(size_t)n * HH + h] - mh);
  red[tid] = s; __syncthreads();
  for (int o = 128; o > 0; o >>= 1) { if (tid < o) red[tid] += red[tid + o]; __syncthreads(); }
  if (tid == 0) { hmax[h] = mh; hsuminv[h] = 1.f / red[0]; }
}

// ---------------------------------------------------------------- pass 2:
// wsum[h][d] = sum_n attn[h][n] * M[n][d]; WMMA partials per (chunk,stripe).
// Branch-free padding: lanes with j>=8 read scores at (j&7) but carry ih==0.
__global__ void k_wsum_part(const float* __restrict__ M,
                            const float* __restrict__ scores,
                            const float* __restrict__ hmax,
                            const float* __restrict__ hsuminv,
                            float* __restrict__ part) {
  int c = blockIdx.x;                        // column chunk (32)
  int tid = threadIdx.x;                     // 256 = 8 waves
  int wave = tid >> 5, lane = tid & 31;
  int s = blockIdx.y * 8 + wave;             // stripe, wave-uniform
  if (s >= NSTRIPE) return;                  // uniform exit before any WMMA
  int d0 = c * 16;
  int j  = lane & 15;
  int jc = j & 7;                            // safe score/head index
  int ks = (lane < 16) ? 0 : 2;
  float mh = hmax[jc];
  float ih = (j < HH) ? hsuminv[jc] : 0.f;   // loop-invariant zero for pad lanes
  v8f acc = {};
  int nbeg = s * ROWS_PER_STRIPE, nend = nbeg + ROWS_PER_STRIPE;
  for (int n0 = nbeg; n0 < nend; n0 += 4) {
    v2f a, b;
    a.x = __expf(scores[(size_t)(n0 + ks)     * HH + jc] - mh) * ih;
    a.y = __expf(scores[(size_t)(n0 + ks + 1) * HH + jc] - mh) * ih;
    b.x = M[(size_t)(n0 + ks)     * DD + d0 + j];   // B[kk][d]
    b.y = M[(size_t)(n0 + ks + 1) * DD + d0 + j];
    acc = __builtin_amdgcn_wmma_f32_16x16x4_f32(false, a, false, b,
                                                (short)0, acc, false, false);
  }
  float* p = part + (size_t)(c * NSTRIPE + s) * 256;
#pragma unroll
  for (int r = 0; r < 8; ++r) p[r * 32 + lane] = acc[r];
}

__global__ void k_wsum_reduce(const float* __restrict__ part,
                              float* __restrict__ wsum) {
  int c = blockIdx.x, e = threadIdx.x;       // 32 blocks x 256 threads
  float s = 0.f;
  for (int st = 0; st < NSTRIPE; ++st) s += part[(size_t)(c * NSTRIPE + st) * 256 + e];
  int r = e >> 5, l = e & 31;
  int h = (l < 16) ? r : r + 8;
  int d = c * 16 + (l & 15);
  if (h < HH) wsum[h * DD + d] = s;
}

// ---------------------------------------------------------------- ctx + out-proj + norm
__global__ void k_proj(const float* __restrict__ inW,
                       const float* __restrict__ inB,
                       const float* __restrict__ outW,
                       const float* __restrict__ outB,
                       const float* __restrict__ wsum,
                       float* __restrict__ attn_out,
                       float* __restrict__ out_norm_inv) {
  __shared__ float ctx[DD];
  __shared__ float red[DD];
  int i = threadIdx.x;                       // 512 threads
  int h = i >> 6;
  const float* wv = inW + (size_t)(2 * DD + i) * DD;  // Wv row i
  const float* wh = wsum + h * DD;
  float s = inB[2 * DD + i];
  for (int d = 0; d < DD; ++d) s += wv[d] * wh[d];
  ctx[i] = s;
  __syncthreads();
  const float* wo = outW + (size_t)i * DD;
  float o = outB[i];
  for (int d = 0; d < DD; ++d) o += wo[d] * ctx[d];
  attn_out[i] = o;
  red[i] = o * o;
  __syncthreads();
  for (int off = 256; off > 0; off >>= 1) { if (i < off) red[i] += red[i + off]; __syncthreads(); }
  if (i == 0) out_norm_inv[0] = 1.f / fmaxf(sqrtf(red[0]), 1e-8f);
}

// ---------------------------------------------------------------- pass 3: sims
__global__ void k_sims(const float* __restrict__ M,
                       const float* __restrict__ attn_out,
                       const float* __restrict__ sqnorm,
                       const float* __restrict__ out_norm_inv,
                       float* __restrict__ sims) {
  __shared__ float ao[DD];
  int tid = threadIdx.x;                     // 256 = 8 waves, 1 row per wave
  for (int e = tid; e < DD; e += 256) ao[e] = attn_out[e];
  __syncthreads();
  int wave = tid >> 5, lane = tid & 31;
  size_t n = (size_t)blockIdx.x * 8 + wave;
  const float* row = M + n * DD;
  float s = 0.f;
  for (int e = lane; e < DD; e += 32) s += row[e] * ao[e];
  for (int o = 16; o > 0; o >>= 1) s += __shfl_xor(s, o, 32);
  if (lane == 0) {
    float bn = fmaxf(sqrtf(sqnorm[n]), 1e-8f);
    sims[n] = (s / bn) * out_norm_inv[0];
  }
}

// ---------------------------------------------------------------- top-5
__global__ void k_topk(const float* __restrict__ sims,
                       float* __restrict__ out_vals,
                       float* __restrict__ out_idx) {
  __shared__ float sv[256 * 5];
  __shared__ int   si[256 * 5];
  int tid = threadIdx.x;                     // 256 threads
  float v[5]; int ix[5];
#pragma unroll
  for (int t = 0; t < 5; ++t) { v[t] = -3.4e38f; ix[t] = -1; }
  for (int n = tid; n < CAPN; n += 256) {
    float x = sims[n];
    if (x > v[4]) {
      int p = 4;
      while (p > 0 && x > v[p - 1]) { v[p] = v[p - 1]; ix[p] = ix[p - 1]; --p; }
      v[p] = x; ix[p] = n;
    }
  }
#pragma unroll
  for (int t = 0; t < 5; ++t) { sv[tid * 5 + t] = v[t]; si[tid * 5 + t] = ix[t]; }
  __syncthreads();
  for (int off = 128; off > 0; off >>= 1) {
    if (tid < off) {
      float a[5]; int ai[5];
      int pa = 0, pb = 0;
#pragma unroll
      for (int t = 0; t < 5; ++t) {
        float va = sv[tid * 5 + pa], vb = sv[(tid + off) * 5 + pb];
        if (va >= vb) { a[t] = va; ai[t] = si[tid * 5 + pa]; ++pa; }
        else          { a[t] = vb; ai[t] = si[(tid + off) * 5 + pb]; ++pb; }
      }
#pragma unroll
      for (int t = 0; t < 5; ++t) { sv[tid * 5 + t] = a[t]; si[tid * 5 + t] = ai[t]; }
    }
    __syncthreads();
  }
  if (tid < 5) { out_vals[tid] = sv[tid]; out_idx[tid] = (float)si[tid]; }
}

// ================================================================ launch
extern "C" void kernel_launch(void* const* d_in, const int* in_sizes, int n_in,
                              void* d_out, int out_size, void* d_ws, size_t ws_size,
                              hipStream_t stream) {
  const float* query = (const float*)d_in[0];
  const float* M     = (const float*)d_in[1];   // (200000, 512)
  const float* inW   = (const float*)d_in[2];   // (1536, 512)
  const float* inB   = (const float*)d_in[3];   // (1536,)
  const float* outW  = (const float*)d_in[4];   // (512, 512)
  const float* outB  = (const float*)d_in[5];   // (512,)
  // top_k (d_in[6]) is 5 per setup_inputs

  float* ws = (float*)d_ws;
  float* qk      = ws;                 // 4096
  float* qconst  = ws + 4096;          // 8
  float* hmax    = ws + 4104;          // 8
  float* hsuminv = ws + 4112;          // 8
  float* wsum    = ws + 4128;          // 4096
  float* oninv   = ws + 8224;          // 1
  float* scores  = ws + 8448;          // 1,600,000
  float* sqnorm  = ws + 1608448;       // 200,000
  float* part    = ws + 1808448;       // 32*125*256 = 1,024,000

  float* out_attn = (float*)d_out;             // 512
  float* out_sims = out_attn + DD;             // 200000
  float* out_vals = out_sims + CAPN;           // 5
  float* out_idx  = out_vals + 5;              // 5 (indices as float)

  k_prep<<<1, 512, 0, stream>>>(query, inW, inB, qk, qconst);
  k_scores<<<(CAPN / 16 + 7) / 8, 256, 0, stream>>>(M, qk, qconst, scores, sqnorm);
  k_stats<<<HH, 256, 0, stream>>>(scores, hmax, hsuminv);
  k_wsum_part<<<dim3(32, (NSTRIPE + 7) / 8), 256, 0, stream>>>(M, scores, hmax, hsuminv, part);
  k_wsum_reduce<<<32, 256, 0, stream>>>(part, wsum);
  k_proj<<<1, 512, 0, stream>>>(inW, inB, outW, outB, wsum, out_attn, oninv);
  k_sims<<<CAPN / 8, 256, 0, stream>>>(M, out_attn, sqnorm, oninv, out_sims);
  k_topk<<<1, 256, 0, stream>>>(out_sims, out_vals, out_idx);
}